// GraphEncoder_62113817034944
// MI455X (gfx1250) — compile-verified
//
#include <hip/hip_runtime.h>

typedef __attribute__((ext_vector_type(2))) float v2f;
typedef __attribute__((ext_vector_type(8))) float v8f;

#define NNODES 100000
#define NEDGES 1600000
#define NGRAPH 1000
#define MAXDEG 512
// 1/sqrt(1 + 1e-5) for eval-mode BN with running stats (0,1)
#define BN_INV 0.9999950000374997f

// ---------------------------------------------------------------- utilities
__global__ void zero_kernel(float* __restrict__ p, int n) {
  int i = blockIdx.x * blockDim.x + threadIdx.x;
  if (i < n) p[i] = 0.f;
}

__global__ void degree_kernel(const int* __restrict__ dst, int* __restrict__ deg, int e) {
  int i = blockIdx.x * blockDim.x + threadIdx.x;
  if (i < e) atomicAdd(&deg[dst[i]], 1);
}

// h0[n,64] = concat(pos[32], deg_emb[16], seed[1], zero-pad[15]); fused pool into pooled0
__global__ void h0_kernel(const float* __restrict__ pos, const int* __restrict__ seed,
                          const int* __restrict__ deg, const float* __restrict__ deg_emb,
                          const int* __restrict__ graph_id,
                          float* __restrict__ h, float* __restrict__ pooled0, int n) {
  int t = blockIdx.x * blockDim.x + threadIdx.x;
  if (t >= n * 64) return;
  int node = t >> 6, f = t & 63;
  float v = 0.f;
  if (f < 32) {
    v = pos[node * 32 + f];
  } else if (f < 48) {
    int d = deg[node];
    if (d > MAXDEG) d = MAXDEG;
    v = deg_emb[d * 16 + (f - 32)];
  } else if (f == 48) {
    v = (float)seed[node];
  }
  h[t] = v;
  if (f < 49) atomicAdd(&pooled0[graph_id[node] * 64 + f], v);
}

__global__ void copy4_kernel(const float4* __restrict__ s, float4* __restrict__ d, int n4) {
  int i = blockIdx.x * blockDim.x + threadIdx.x;
  if (i < n4) d[i] = s[i];
}

// agg[dst] += h[src]  (16 threads/edge, 4 floats each)
__global__ void scatter_kernel(const int* __restrict__ src, const int* __restrict__ dst,
                               const float* __restrict__ h, float* __restrict__ agg, int e) {
  int t = blockIdx.x * blockDim.x + threadIdx.x;
  if (t >= e * 16) return;
  int ed = t >> 4;
  int fq = (t & 15) << 2;
  int s = src[ed], d = dst[ed];
  float4 v = *(const float4*)&h[s * 64 + fq];
  float* ap = &agg[d * 64 + fq];
  atomicAdd(ap + 0, v.x);
  atomicAdd(ap + 1, v.y);
  atomicAdd(ap + 2, v.z);
  atomicAdd(ap + 3, v.w);
}

// ------------------------------------------------------------ GIN MLP (WMMA)
// One wave = 16-node tile.  h_out = relu(bn3(relu(bn2( relu(bn1(agg@W1+b1)) @W2 + b2 ))))
// A tiles + transposed weights in LDS (stride 66 rows -> conflict-free ds_read_b64).
__launch_bounds__(256)
__global__ void gin_mlp_kernel(const float* __restrict__ agg, float* __restrict__ hout,
                               const float* __restrict__ w1, const float* __restrict__ b1,
                               const float* __restrict__ g1, const float* __restrict__ be1,
                               const float* __restrict__ w2, const float* __restrict__ b2,
                               const float* __restrict__ g2, const float* __restrict__ be2,
                               const float* __restrict__ g3, const float* __restrict__ be3,
                               const int* __restrict__ graph_id, float* __restrict__ pooled,
                               int kdim, int nnodes) {
  __shared__ float sWT[64 * 66];      // W^T, padded rows
  __shared__ float sA[8 * 16 * 66];   // per-wave 16x64 A tiles
  const int tid = threadIdx.x;
  const int lane = tid & 31, wave = tid >> 5;
  const int node0 = blockIdx.x * 128 + wave * 16;
  float* A = &sA[wave * (16 * 66)];
  const int mrow = lane & 15;
  const int kk = (lane >> 4) << 1;       // 0 or 2
  const int mbase = (lane >> 4) << 3;    // 0 or 8 (C/D row group)

  // stage W1^T (zero-pad K rows >= kdim)
  for (int idx = tid; idx < 64 * 64; idx += 256) {
    int k = idx >> 6, nn = idx & 63;
    sWT[nn * 66 + k] = (k < kdim) ? w1[k * 64 + nn] : 0.f;
  }
  // stage A tile (zero-pad past-N rows)
  for (int i = 0; i < 8; ++i) {
    int idx = lane + 32 * i;
    int m = idx >> 4, c = (idx & 15) << 2;
    int node = node0 + m;
    float4 v = make_float4(0.f, 0.f, 0.f, 0.f);
    if (node < nnodes) v = *(const float4*)&agg[node * 64 + c];
    float* ap = &A[m * 66 + c];
    ap[0] = v.x; ap[1] = v.y; ap[2] = v.z; ap[3] = v.w;
  }
  __syncthreads();

  v8f acc[4] = {};
#pragma unroll
  for (int k0 = 0; k0 < 64; k0 += 4) {
    v2f a = *(const v2f*)&A[mrow * 66 + k0 + kk];
#pragma unroll
    for (int ct = 0; ct < 4; ++ct) {
      v2f b = *(const v2f*)&sWT[(mrow + 16 * ct) * 66 + k0 + kk];
      acc[ct] = __builtin_amdgcn_wmma_f32_16x16x4_f32(false, a, false, b, (short)0,
                                                      acc[ct], false, false);
    }
  }
  __syncthreads();  // all waves done reading W1^T

  // epilogue 1: bias + BN + ReLU, write t1 back into own A tile
#pragma unroll
  for (int ct = 0; ct < 4; ++ct) {
    int n = mrow + 16 * ct;
    float bb = b1[n], gg = g1[n] * BN_INV, be = be1[n];
#pragma unroll
    for (int r = 0; r < 8; ++r) {
      float t = (acc[ct][r] + bb) * gg + be;
      A[(r + mbase) * 66 + n] = fmaxf(t, 0.f);
    }
  }
  // stage W2^T
  for (int idx = tid; idx < 64 * 64; idx += 256) {
    int k = idx >> 6, nn = idx & 63;
    sWT[nn * 66 + k] = w2[k * 64 + nn];
  }
  __syncthreads();

  v8f acc2[4] = {};
#pragma unroll
  for (int k0 = 0; k0 < 64; k0 += 4) {
    v2f a = *(const v2f*)&A[mrow * 66 + k0 + kk];
#pragma unroll
    for (int ct = 0; ct < 4; ++ct) {
      v2f b = *(const v2f*)&sWT[(mrow + 16 * ct) * 66 + k0 + kk];
      acc2[ct] = __builtin_amdgcn_wmma_f32_16x16x4_f32(false, a, false, b, (short)0,
                                                       acc2[ct], false, false);
    }
  }

  // epilogue 2: bias + BN2/ReLU + BN3/ReLU, store h and fused graph pooling
#pragma unroll
  for (int ct = 0; ct < 4; ++ct) {
    int n = mrow + 16 * ct;
    float bb = b2[n];
    float g2n = g2[n] * BN_INV, be2n = be2[n];
    float g3n = g3[n] * BN_INV, be3n = be3[n];
#pragma unroll
    for (int r = 0; r < 8; ++r) {
      int node = node0 + r + mbase;
      float t = acc2[ct][r] + bb;
      t = fmaxf(g2n * t + be2n, 0.f);
      t = fmaxf(g3n * t + be3n, 0.f);
      if (node < nnodes) {
        hout[node * 64 + n] = t;
        atomicAdd(&pooled[graph_id[node] * 64 + n], t);
      }
    }
  }
}

// ------------------------------------------------------- JK readout (WMMA)
// score[g] = sum_i pooled_i[g] @ W_i + sum_i b_i
__launch_bounds__(256)
__global__ void score_kernel(const float* __restrict__ pooled,
                             const float* __restrict__ w0, const float* __restrict__ w1,
                             const float* __restrict__ w2, const float* __restrict__ w3,
                             const float* __restrict__ w4,
                             const float* __restrict__ bb0, const float* __restrict__ bb1,
                             const float* __restrict__ bb2, const float* __restrict__ bb3,
                             const float* __restrict__ bb4,
                             float* __restrict__ out, int g) {
  __shared__ float sWT[64 * 66];
  __shared__ float sA[8 * 16 * 66];
  const int tid = threadIdx.x;
  const int lane = tid & 31, wave = tid >> 5;
  const int g0 = blockIdx.x * 128 + wave * 16;
  float* A = &sA[wave * (16 * 66)];
  const int mrow = lane & 15;
  const int kk = (lane >> 4) << 1;
  const int mbase = (lane >> 4) << 3;

  const float* wlist[5] = {w0, w1, w2, w3, w4};
  v8f acc[4] = {};
#pragma unroll
  for (int li = 0; li < 5; ++li) {
    __syncthreads();  // protect sWT reuse across layers
    const float* w = wlist[li];
    const int kd = (li == 0) ? 49 : 64;
    for (int idx = tid; idx < 64 * 64; idx += 256) {
      int k = idx >> 6, nn = idx & 63;
      sWT[nn * 66 + k] = (k < kd) ? w[k * 64 + nn] : 0.f;
    }
    const float* P = pooled + li * NGRAPH * 64;
    for (int i = 0; i < 8; ++i) {
      int idx = lane + 32 * i;
      int m = idx >> 4, c = (idx & 15) << 2;
      int row = g0 + m;
      float4 v = make_float4(0.f, 0.f, 0.f, 0.f);
      if (row < g) v = *(const float4*)&P[row * 64 + c];
      float* ap = &A[m * 66 + c];
      ap[0] = v.x; ap[1] = v.y; ap[2] = v.z; ap[3] = v.w;
    }
    __syncthreads();
#pragma unroll
    for (int k0 = 0; k0 < 64; k0 += 4) {
      v2f a = *(const v2f*)&A[mrow * 66 + k0 + kk];
#pragma unroll
      for (int ct = 0; ct < 4; ++ct) {
        v2f b = *(const v2f*)&sWT[(mrow + 16 * ct) * 66 + k0 + kk];
        acc[ct] = __builtin_amdgcn_wmma_f32_16x16x4_f32(false, a, false, b, (short)0,
                                                        acc[ct], false, false);
      }
    }
  }
#pragma unroll
  for (int ct = 0; ct < 4; ++ct) {
    int n = mrow + 16 * ct;
    float bsum = bb0[n] + bb1[n] + bb2[n] + bb3[n] + bb4[n];
#pragma unroll
    for (int r = 0; r < 8; ++r) {
      int row = g0 + r + mbase;
      if (row < g) out[row * 64 + n] = acc[ct][r] + bsum;
    }
  }
}

// ---------------------------------------------------------------- launcher
extern "C" void kernel_launch(void* const* d_in, const int* in_sizes, int n_in,
                              void* d_out, int out_size, void* d_ws, size_t ws_size,
                              hipStream_t stream) {
  (void)in_sizes; (void)n_in; (void)out_size; (void)ws_size;
  const float* pos      = (const float*)d_in[0];
  const int*   seed     = (const int*)d_in[1];
  const int*   src      = (const int*)d_in[2];
  const int*   dst      = (const int*)d_in[3];
  const int*   graph_id = (const int*)d_in[4];
  const float* deg_emb  = (const float*)d_in[5];
  auto gin = [&](int L, int j) { return (const float*)d_in[6 + L * 10 + j]; };
  auto pw  = [&](int i) { return (const float*)d_in[46 + i * 2]; };
  auto pb  = [&](int i) { return (const float*)d_in[47 + i * 2]; };

  float* h      = (float*)d_ws;                       // [N,64]
  float* agg    = h + (size_t)NNODES * 64;            // [N,64]
  float* pooled = agg + (size_t)NNODES * 64;          // [5,G,64]
  int*   deg    = (int*)(pooled + 5 * NGRAPH * 64);   // [N]

  zero_kernel<<<(5 * NGRAPH * 64 + 255) / 256, 256, 0, stream>>>(pooled, 5 * NGRAPH * 64);
  zero_kernel<<<(NNODES + 255) / 256, 256, 0, stream>>>((float*)deg, NNODES);
  degree_kernel<<<(NEDGES + 255) / 256, 256, 0, stream>>>(dst, deg, NEDGES);
  h0_kernel<<<(NNODES * 64 + 255) / 256, 256, 0, stream>>>(pos, seed, deg, deg_emb,
                                                           graph_id, h, pooled, NNODES);

  const int mlp_blocks = (NNODES + 127) / 128;
  for (int L = 0; L < 4; ++L) {
    copy4_kernel<<<(NNODES * 16 + 255) / 256, 256, 0, stream>>>((const float4*)h,
                                                                (float4*)agg, NNODES * 16);
    scatter_kernel<<<(NEDGES * 16 + 255) / 256, 256, 0, stream>>>(src, dst, h, agg, NEDGES);
    gin_mlp_kernel<<<mlp_blocks, 256, 0, stream>>>(
        agg, h, gin(L, 0), gin(L, 1), gin(L, 2), gin(L, 3), gin(L, 4), gin(L, 5),
        gin(L, 6), gin(L, 7), gin(L, 8), gin(L, 9), graph_id,
        pooled + (size_t)(L + 1) * NGRAPH * 64, (L == 0) ? 49 : 64, NNODES);
  }
  score_kernel<<<(NGRAPH + 127) / 128, 256, 0, stream>>>(
      pooled, pw(0), pw(1), pw(2), pw(3), pw(4), pb(0), pb(1), pb(2), pb(3), pb(4),
      (float*)d_out, NGRAPH);
}